// SimpleLESS4FDModel_43069932045125
// MI455X (gfx1250) — compile-verified
//
#include <hip/hip_runtime.h>
#include <math.h>

typedef float v2f __attribute__((ext_vector_type(2)));
typedef float v8f __attribute__((ext_vector_type(8)));
typedef unsigned int u32x4 __attribute__((ext_vector_type(4)));
typedef int i32x4 __attribute__((ext_vector_type(4)));
typedef int i32x8 __attribute__((ext_vector_type(8)));

#define HID 64
#define NH  4
#define HD  16

#ifndef USE_TDM
#if defined(__has_builtin)
#if __has_builtin(__builtin_amdgcn_tensor_load_to_lds) && __has_builtin(__builtin_amdgcn_s_wait_tensorcnt)
#define USE_TDM 1
#else
#define USE_TDM 0
#endif
#else
#define USE_TDM 0
#endif
#endif

// ---------- device helpers ----------

__device__ __forceinline__ float gelu_tanh(float x) {
  // jax.nn.gelu default (approximate=True)
  const float c0 = 0.7978845608028654f; // sqrt(2/pi)
  float x3 = x * x * x;
  return 0.5f * x * (1.0f + tanhf(c0 * (x + 0.044715f * x3)));
}

// monotone float<->uint encoding so atomicMax(uint) == float max; 0 acts as -inf identity
__device__ __forceinline__ unsigned encodeF(float f) {
  unsigned u = __float_as_uint(f);
  return ((int)u < 0) ? ~u : (u | 0x80000000u);
}
__device__ __forceinline__ float decodeF(unsigned u) {
  return (u & 0x80000000u) ? __uint_as_float(u & 0x7fffffffu) : __uint_as_float(~u);
}

__device__ __forceinline__ void atomicAddF(float* p, float v) {
  unsafeAtomicAdd(p, v);  // hardware global_atomic_add_f32
}

#if USE_TDM
// Issue a TDM load of one weight K-tile: 64 rows x LD cols of f32, contiguous in
// global memory, written to LDS with an (LD+8)-float padded row stride via the
// TDM pad feature (pad 8 DWORDs after every LD DWORDs). D# layout per CDNA5 ISA
// 8.3/8.4: group0 = {count/type/addr}, group1 = {pads, dims, strides}, 1-D tile.
// This toolchain exposes the 6-arg builtin: (g0 4dw, g1 8dw, g2 4dw, g3 4dw,
// extra 8dw, cpol); groups 2/3 and the extra operand are zero for <=2-D tensors.
__device__ __forceinline__ void tdm_load_wtile(const float* gsrc, unsigned lds_off, int LD) {
  unsigned long long ga = (unsigned long long)(size_t)gsrc;
  unsigned nelem = (unsigned)(64 * LD);               // f32 elements in tile (<= 4096)
  unsigned padInterval = (LD == 64) ? 5u : 4u;        // code: 5 -> 64 dw, 4 -> 32 dw
  unsigned padAmount = 7u;                            // code 7 -> 8 dwords of pad

  u32x4 g0;
  g0.x = 1u;                                          // count=1 (valid user descriptor)
  g0.y = lds_off;                                     // lds_addr (bytes)
  g0.z = (unsigned)ga;                                // global_addr[31:0]
  g0.w = (unsigned)((ga >> 32) & 0x01ffffffull) | (2u << 30);  // addr[56:32] | type=2

  i32x8 g1;
  g1[0] = (int)((2u << 16) | (1u << 20) | (padInterval << 22) | (padAmount << 25)); // data_size=4B, pad_enable
  g1[1] = (int)((nelem & 0xffffu) << 16);             // tensor_dim0[15:0] -> bits 63:48
  g1[2] = (int)((nelem >> 16) | (1u << 16));          // tensor_dim0[31:16] | tensor_dim1=1
  g1[3] = (int)((nelem & 0xffffu) << 16);             // tile_dim0 -> bits 127:112
  g1[4] = 0;                                          // tile_dim1=0, tile_dim2=0 (1-D tile)
  g1[5] = (int)nelem;                                 // tensor_dim0_stride[31:0]
  g1[6] = 0;
  g1[7] = 0;

  i32x4 gz4 = {0, 0, 0, 0};                           // groups 2/3 unused (<=2D)
  i32x8 gz8 = {0, 0, 0, 0, 0, 0, 0, 0};               // extra operand (zero-filled)
  __builtin_amdgcn_tensor_load_to_lds(g0, g1, gz4, gz4, gz8, 0);
}
#endif

// ---------- WMMA fp32 GEMM:  Out[N x (COLT*16)] = op(A[N x K]) @ W[K x COLT*16] + bias ----------
// One wave computes a 16-row x (COLT*16)-col tile using V_WMMA_F32_16X16X4_F32.
// W is staged K-tile-by-K-tile (64 rows) into LDS (TDM if available, else cooperative
// b128 loads) with padded stride so the two half-wave B-fragment reads are bank-conflict-free.
// IN_MODE : 0 = plain A load ; 1 = A := gelu(A / (denom[row,head]+1e-16))  (HGT agg normalize)
// OUT_MODE: 0 = +bias ; 1 = +bias, skip-blend with sigmoid(skipLogit), then relu ; 2 = +bias, relu
template <int COLT, int IN_MODE, int OUT_MODE>
__global__ __launch_bounds__(256)
void gemm_wmma_kernel(const float* __restrict__ A,
                      const float* __restrict__ W,
                      const float* __restrict__ bias,
                      float* __restrict__ Out,
                      const float* __restrict__ denom,
                      const float* __restrict__ skipPrev,
                      const float* __restrict__ skipLogit,
                      int N, int K) {
  const int LD = COLT * 16;       // W / Out leading dimension
  const int SW = LD + 8;          // padded LDS row stride (floats)
  const int BK = 64;              // K-tile depth staged per iteration
  __shared__ float sW[(COLT * 16 + 8) * 64];

  const int wave = (int)(((size_t)blockIdx.x * blockDim.x + threadIdx.x) >> 5);
  const int lane = threadIdx.x & 31;
  const int ntiles = N >> 4;
  const bool active = wave < ntiles;

  const int row0 = (active ? wave : 0) << 4;
  const bool hi  = lane >= 16;
  const int l16  = lane & 15;
  const int arow = row0 + l16;       // A-matrix: lanes 0-15 / 16-31 both hold rows M=0..15
  const int koff = hi ? 2 : 0;       // A vgpr0/1 hold K={0,1} (lo lanes) or K={2,3} (hi lanes)

  v8f acc[COLT];
#pragma unroll
  for (int t = 0; t < COLT; ++t) acc[t] = v8f{};

  const float* __restrict__ abase = A + (size_t)arow * K;

  for (int k0 = 0; k0 < K; k0 += BK) {
    __syncthreads();  // previous tile fully consumed
#if USE_TDM
    if (threadIdx.x == 0) {
      tdm_load_wtile(W + (size_t)k0 * LD, (unsigned)(size_t)&sW[0], LD);
      __builtin_amdgcn_s_wait_tensorcnt(0);
    }
#else
    for (int idx = threadIdx.x; idx < BK * (LD / 4); idx += 256) {
      const int r = idx / (LD / 4);
      const int c4 = idx - r * (LD / 4);
      const float4 v = *(const float4*)(W + (size_t)(k0 + r) * LD + c4 * 4);
      *(float4*)(&sW[r * SW + c4 * 4]) = v;
    }
#endif
    __syncthreads();

    for (int k = 0; k < BK; k += 4) {
      const int kk = k + koff;       // tile-local row for this half-wave
      v2f a;
      if (IN_MODE == 0) {
        a.x = abase[k0 + kk];
        a.y = abase[k0 + kk + 1];
      } else {
        // kk and kk+1 always lie inside the same 16-wide head block
        const float d = denom[(size_t)arow * NH + ((k0 + kk) >> 4)] + 1e-16f;
        a.x = gelu_tanh(abase[k0 + kk] / d);
        a.y = gelu_tanh(abase[k0 + kk + 1] / d);
      }
      const float* wrow = &sW[kk * SW + l16];
#pragma unroll
      for (int t = 0; t < COLT; ++t) {
        v2f b;
        b.x = wrow[t * 16];        // B row kk (lo) / kk+2 (hi)
        b.y = wrow[t * 16 + SW];   // B row kk+1 (lo) / kk+3 (hi)
        acc[t] = __builtin_amdgcn_wmma_f32_16x16x4_f32(
            false, a, false, b, (short)0, acc[t], false, false);
      }
    }
  }

  if (!active) return;

  float g = 0.0f;
  if (OUT_MODE == 1) g = 1.0f / (1.0f + __expf(-skipLogit[0]));

#pragma unroll
  for (int t = 0; t < COLT; ++t) {
    const int col = t * 16 + l16;
    const float bcol = bias[col];
#pragma unroll
    for (int i = 0; i < 8; ++i) {
      // C/D layout: vgpr i holds row i (lanes 0-15) or row i+8 (lanes 16-31)
      const int row = row0 + i + (hi ? 8 : 0);
      float o = acc[t][i] + bcol;
      if (OUT_MODE == 1) {
        o = g * o + (1.0f - g) * skipPrev[(size_t)row * HID + col];
        o = fmaxf(o, 0.0f);
      } else if (OUT_MODE == 2) {
        o = fmaxf(o, 0.0f);
      }
      Out[(size_t)row * LD + col] = o;
    }
  }
}

// ---------- fuse per-head relation matrix into projection weights ----------
// Wdst[i, h*16+f] = sum_d Wsrc[i, h*16+d] * rel[h, d, f];  row i==64 handles the bias.
__global__ void fuse_rel_kernel(const float* __restrict__ Wsrc, const float* __restrict__ bsrc,
                                const float* __restrict__ rel,
                                float* __restrict__ Wdst, float* __restrict__ bdst) {
  int idx = blockIdx.x * blockDim.x + threadIdx.x;
  if (idx >= 65 * 64) return;
  int i = idx >> 6, j = idx & 63;
  int hh = j >> 4, f = j & 15;
  const float* rp = rel + hh * 256 + f;  // rel[h, d, f], stride 16 over d
  float s = 0.0f;
  if (i < 64) {
    const float* wp = Wsrc + i * 64 + hh * 16;
#pragma unroll
    for (int d = 0; d < 16; ++d) s += wp[d] * rp[d * 16];
    Wdst[i * 64 + j] = s;
  } else {
    const float* bp = bsrc + hh * 16;
#pragma unroll
    for (int d = 0; d < 16; ++d) s += bp[d] * rp[d * 16];
    bdst[j] = s;
  }
}

// ---------- edge pass 1: attention logits + segment max (thread = edge*head) ----------
__global__ void edge_alpha_kernel(const int* __restrict__ ei,
                                  const float* __restrict__ ka, const float* __restrict__ q,
                                  const float* __restrict__ prel,
                                  float* __restrict__ alpha, unsigned* __restrict__ mx, int E_) {
  long long t = (long long)blockIdx.x * blockDim.x + threadIdx.x;
  if (t >= (long long)E_ * NH) return;
  int e = (int)(t >> 2), hh = (int)(t & 3);
  int src = ei[e], dst = ei[E_ + e];
  const float4* kp = (const float4*)(ka + (size_t)src * HID + hh * HD);
  const float4* qp = (const float4*)(q + (size_t)dst * HID + hh * HD);
  float s = 0.0f;
#pragma unroll
  for (int j = 0; j < 4; ++j) {
    float4 a4 = kp[j], b4 = qp[j];
    s += a4.x * b4.x + a4.y * b4.y + a4.z * b4.z + a4.w * b4.w;
  }
  s *= prel[hh] * 0.25f;  // 1/sqrt(D), D=16
  alpha[(size_t)e * NH + hh] = s;
  atomicMax(&mx[(size_t)dst * NH + hh], encodeF(s));
}

// ---------- edge pass 2: exp + scatter numerator/denominator (16 lanes per edge) ----------
__global__ void edge_agg_kernel(const int* __restrict__ ei,
                                const float* __restrict__ alpha, const unsigned* __restrict__ mx,
                                const float* __restrict__ vm,
                                float* __restrict__ denom, float* __restrict__ agg, int E_) {
  long long t = (long long)blockIdx.x * blockDim.x + threadIdx.x;
  if (t >= (long long)E_ * 16) return;
  int e = (int)(t >> 4), i = (int)(t & 15);
  int hh = i >> 2;
  int src = ei[e], dst = ei[E_ + e];
  float m = decodeF(mx[(size_t)dst * NH + hh]);
  float a = __expf(alpha[(size_t)e * NH + hh] - m);
  if ((i & 3) == 0) atomicAddF(&denom[(size_t)dst * NH + hh], a);
  float4 v4 = *(const float4*)(vm + (size_t)src * HID + i * 4);
  float* ap = agg + (size_t)dst * HID + i * 4;
  atomicAddF(ap + 0, a * v4.x);
  atomicAddF(ap + 1, a * v4.y);
  atomicAddF(ap + 2, a * v4.z);
  atomicAddF(ap + 3, a * v4.w);
}

// ---------- final tiny projection: logits[N,2] = hc[N,32] @ Wc2 + bc2 ----------
__global__ void logits_kernel(const float* __restrict__ hc,
                              const float* __restrict__ Wc2, const float* __restrict__ bc2,
                              float* __restrict__ out, int N_) {
  int n = blockIdx.x * blockDim.x + threadIdx.x;
  if (n >= N_) return;
  const float4* hp = (const float4*)(hc + (size_t)n * 32);
  float o0 = bc2[0], o1 = bc2[1];
#pragma unroll
  for (int j4 = 0; j4 < 8; ++j4) {
    float4 h4 = hp[j4];
    int j = j4 * 4;
    o0 += h4.x * Wc2[(j + 0) * 2] + h4.y * Wc2[(j + 1) * 2] + h4.z * Wc2[(j + 2) * 2] + h4.w * Wc2[(j + 3) * 2];
    o1 += h4.x * Wc2[(j + 0) * 2 + 1] + h4.y * Wc2[(j + 1) * 2 + 1] + h4.z * Wc2[(j + 2) * 2 + 1] + h4.w * Wc2[(j + 3) * 2 + 1];
  }
  out[(size_t)n * 2 + 0] = o0;
  out[(size_t)n * 2 + 1] = o1;
}

// ---------- host launch ----------

extern "C" void kernel_launch(void* const* d_in, const int* in_sizes, int n_in,
                              void* d_out, int out_size, void* d_ws, size_t ws_size,
                              hipStream_t stream) {
  const float* x     = (const float*)d_in[0];
  const int*   ei    = (const int*)d_in[1];
  const float* Wp    = (const float*)d_in[2];
  const float* bp    = (const float*)d_in[3];
  const float* Wk    = (const float*)d_in[4];
  const float* bk    = (const float*)d_in[5];
  const float* Wq    = (const float*)d_in[6];
  const float* bq    = (const float*)d_in[7];
  const float* Wv    = (const float*)d_in[8];
  const float* bv    = (const float*)d_in[9];
  const float* a_rel = (const float*)d_in[10];
  const float* m_rel = (const float*)d_in[11];
  const float* p_rel = (const float*)d_in[12];
  const float* Wo    = (const float*)d_in[13];
  const float* bo    = (const float*)d_in[14];
  const float* skip  = (const float*)d_in[15];
  const float* Wc1   = (const float*)d_in[16];
  const float* bc1   = (const float*)d_in[17];
  const float* Wc2   = (const float*)d_in[18];
  const float* bc2   = (const float*)d_in[19];

  const int FIN = 768, L = 2;
  const int N = in_sizes[0] / FIN;
  const int E = in_sizes[1] / 2;

  char* ws = (char*)d_ws;
  size_t o = 0;
  auto take = [&](size_t bytes) { size_t r = o; o += (bytes + 255) & ~(size_t)255; return r; };
  float*    h0    = (float*)(ws + take((size_t)N * HID * 4));
  float*    h1    = (float*)(ws + take((size_t)N * HID * 4));
  float*    qb    = (float*)(ws + take((size_t)N * HID * 4));
  float*    kab   = (float*)(ws + take((size_t)N * HID * 4));
  float*    vmb   = (float*)(ws + take((size_t)N * HID * 4));
  float*    hcb   = (float*)(ws + take((size_t)N * 32 * 4));
  float*    alpha = (float*)(ws + take((size_t)E * NH * 4));
  unsigned* mx    = (unsigned*)(ws + take((size_t)N * NH * 4));
  float*    denom = (float*)(ws + take((size_t)N * NH * 4));
  float*    agg   = (float*)(ws + take((size_t)N * HID * 4));
  float*    Wka   = (float*)(ws + take(64 * 64 * 4));
  float*    bka   = (float*)(ws + take(64 * 4));
  float*    Wvm   = (float*)(ws + take(64 * 64 * 4));
  float*    bvm   = (float*)(ws + take(64 * 4));
  (void)ws_size; (void)n_in; (void)out_size;

  const int TB = 256;
  const int wavesPerBlk = TB / 32;
  const int gemmBlocks = ((N >> 4) + wavesPerBlk - 1) / wavesPerBlk;

  // h0 = x @ Wp + bp   (bandwidth-bound over x; fp32 WMMA, K=768, W staged 64 rows at a time)
  gemm_wmma_kernel<4, 0, 0><<<gemmBlocks, TB, 0, stream>>>(
      x, Wp, bp, h0, nullptr, nullptr, nullptr, N, FIN);

  float* hcur = h0;
  float* hnxt = h1;
  for (int l = 0; l < L; ++l) {
    // Fold a_rel/m_rel (single edge type) into node-side weights: ka = h@(Wk A)+bk A, vm = h@(Wv M)+bv M
    fuse_rel_kernel<<<17, TB, 0, stream>>>(Wk + l * 4096, bk + l * 64, a_rel + l * 1024, Wka, bka);
    fuse_rel_kernel<<<17, TB, 0, stream>>>(Wv + l * 4096, bv + l * 64, m_rel + l * 1024, Wvm, bvm);

    gemm_wmma_kernel<4, 0, 0><<<gemmBlocks, TB, 0, stream>>>(
        hcur, Wq + l * 4096, bq + l * 64, qb, nullptr, nullptr, nullptr, N, HID);
    gemm_wmma_kernel<4, 0, 0><<<gemmBlocks, TB, 0, stream>>>(
        hcur, Wka, bka, kab, nullptr, nullptr, nullptr, N, HID);
    gemm_wmma_kernel<4, 0, 0><<<gemmBlocks, TB, 0, stream>>>(
        hcur, Wvm, bvm, vmb, nullptr, nullptr, nullptr, N, HID);

    (void)hipMemsetAsync(mx, 0, (size_t)N * NH * 4, stream);
    (void)hipMemsetAsync(denom, 0, (size_t)N * NH * 4, stream);
    (void)hipMemsetAsync(agg, 0, (size_t)N * HID * 4, stream);

    long long t1 = (long long)E * NH;
    edge_alpha_kernel<<<(unsigned)((t1 + TB - 1) / TB), TB, 0, stream>>>(
        ei, kab, qb, p_rel + l * NH, alpha, mx, E);
    long long t2 = (long long)E * 16;
    edge_agg_kernel<<<(unsigned)((t2 + TB - 1) / TB), TB, 0, stream>>>(
        ei, alpha, mx, vmb, denom, agg, E);

    // h_next = relu( g*(gelu(agg/denom) @ Wo + bo) + (1-g)*h )   [norm+gelu fused into A-load]
    gemm_wmma_kernel<4, 1, 1><<<gemmBlocks, TB, 0, stream>>>(
        agg, Wo + l * 4096, bo + l * 64, hnxt, denom, hcur, skip + l, N, HID);

    float* tmp = hcur; hcur = hnxt; hnxt = tmp;
  }

  // classifier: hc = relu(h @ Wc1 + bc1)  (WMMA, 2 column tiles), then logits = hc @ Wc2 + bc2
  gemm_wmma_kernel<2, 0, 2><<<gemmBlocks, TB, 0, stream>>>(
      hcur, Wc1, bc1, hcb, nullptr, nullptr, nullptr, N, HID);
  logits_kernel<<<(N + TB - 1) / TB, TB, 0, stream>>>(hcb, Wc2, bc2, (float*)d_out, N);
}